// PointPillarScatter_24206435680687
// MI455X (gfx1250) — compile-verified
//
#include <hip/hip_runtime.h>
#include <stdint.h>

// PointPillarScatter for MI455X (gfx1250).
// Strategy: invert the scatter into a gather so ALL 219MB of output writes are
// coalesced b128 non-temporal stores.
//   K1: init cell->pillar map (d_ws) to -1            (3.4 MB NT b128 stores)
//   K2: scatter pillar ids into the map               (48K 4B stores)
//   K3: gather: 1 thread per 4 cells, loop 64 channels; predicated b128 row
//       reads + 4x4 register transpose + NT b128 stores.

#define NX_    432
#define NY_    496
#define C_     64
#define PLANE_ (NX_ * NY_)          // 214272 (divisible by 4 and 256)
#define BATCH_ 4
#define CELLS_ (BATCH_ * PLANE_)    // 857088

typedef float v4f __attribute__((ext_vector_type(4)));
typedef int   v4i __attribute__((ext_vector_type(4)));

// Address-space-qualified pointer types matching the async-LDS builtin's
// signature (param 0 printed by clang as 'int __vector(4) __device__ *').
typedef __attribute__((address_space(1))) v4i* as1_v4i_ptr;
typedef __attribute__((address_space(3))) v4i* as3_v4i_ptr;

#if defined(__has_builtin) && __has_builtin(__builtin_amdgcn_global_load_async_to_lds_b128)
#define PPS_ASYNC_LDS 1
#else
#define PPS_ASYNC_LDS 0
#endif

__global__ void ppscatter_init_map(int* __restrict__ map) {
    const int t = blockIdx.x * blockDim.x + threadIdx.x;   // one thread per 4 cells
    if (t < CELLS_ / 4) {
        v4i m1 = {-1, -1, -1, -1};
        __builtin_nontemporal_store(m1, ((v4i*)map) + t);
    }
}

__global__ void ppscatter_ids(const int* __restrict__ coords,
                              int* __restrict__ map, int npillars) {
    const int p = blockIdx.x * blockDim.x + threadIdx.x;
    if (p < npillars) {
        const int b = coords[4 * p + 0];
        const int z = coords[4 * p + 1];   // always 0 (NZ==1), kept to mirror ref
        const int y = coords[4 * p + 2];
        const int x = coords[4 * p + 3];
        map[b * PLANE_ + z + y * NX_ + x] = p;
    }
}

__global__ void ppscatter_gather(const int* __restrict__ map,
                                 const float* __restrict__ feat,
                                 float* __restrict__ out) {
    const int t     = threadIdx.x;
    const int c4    = blockIdx.x * blockDim.x + t;  // 4-cell group index (exact grid)
    const int cell0 = c4 * 4;

#if PPS_ASYNC_LDS
    // Stage this block's map slice through LDS with the CDNA5 async path
    // (tracked by ASYNCcnt) instead of a plain vector load.
    __shared__ int smap[256 * 4];
    {
        const v4i* g = ((const v4i*)map) + c4;
        v4i*       l = ((v4i*)smap) + t;
        __builtin_amdgcn_global_load_async_to_lds_b128(
            (as1_v4i_ptr)(uintptr_t)g,
            (as3_v4i_ptr)(uintptr_t)l,
            /*offset=*/0, /*cpol=*/0);
#if defined(__has_builtin) && __has_builtin(__builtin_amdgcn_s_wait_asynccnt)
        __builtin_amdgcn_s_wait_asynccnt(0);
#else
        asm volatile("s_wait_asynccnt 0" ::: "memory");
#endif
    }
    __syncthreads();
    const v4i pv = ((const v4i*)smap)[t];
#else
    const v4i pv = *(((const v4i*)map) + c4);
#endif

    const int b  = cell0 / PLANE_;        // 4 cells never straddle a batch (PLANE_%4==0)
    const int yx = cell0 - b * PLANE_;
    float* outBase = out + (size_t)b * C_ * PLANE_ + yx;  // + c*PLANE_ ; 16B aligned

    const bool a0 = pv.x >= 0, a1 = pv.y >= 0, a2 = pv.z >= 0, a3 = pv.w >= 0;
    // Clamp so addresses are always valid; masks select zero for empty cells.
    const float* r0 = feat + (size_t)(a0 ? pv.x : 0) * C_;
    const float* r1 = feat + (size_t)(a1 ? pv.y : 0) * C_;
    const float* r2 = feat + (size_t)(a2 ? pv.z : 0) * C_;
    const float* r3 = feat + (size_t)(a3 ? pv.w : 0) * C_;

    // Pull the scattered 256B feature rows toward the WGP (global_prefetch_b8).
    __builtin_prefetch(r0);
    __builtin_prefetch(r1);
    __builtin_prefetch(r2);
    __builtin_prefetch(r3);

    const v4f zero = {0.f, 0.f, 0.f, 0.f};
    for (int cc = 0; cc < C_; cc += 4) {
        v4f f0 = a0 ? *(const v4f*)(r0 + cc) : zero;
        v4f f1 = a1 ? *(const v4f*)(r1 + cc) : zero;
        v4f f2 = a2 ? *(const v4f*)(r2 + cc) : zero;
        v4f f3 = a3 ? *(const v4f*)(r3 + cc) : zero;
#pragma unroll
        for (int j = 0; j < 4; ++j) {
            v4f o = {f0[j], f1[j], f2[j], f3[j]};   // 4x4 transpose in registers
            __builtin_nontemporal_store(o, (v4f*)(outBase + (size_t)(cc + j) * PLANE_));
        }
    }
}

extern "C" void kernel_launch(void* const* d_in, const int* in_sizes, int n_in,
                              void* d_out, int out_size, void* d_ws, size_t ws_size,
                              hipStream_t stream) {
    const float* feat   = (const float*)d_in[0];   // [B*P, 64] fp32
    const int*   coords = (const int*)d_in[1];     // [B*P, 4]  int32
    float*       out    = (float*)d_out;           // [B, 64, NY, NX] fp32
    int*         map    = (int*)d_ws;              // [CELLS_] int32 scratch (3.4 MB)

    const int npillars = in_sizes[0] / C_;         // 48000

    const dim3 blk(256);
    ppscatter_init_map<<<dim3((CELLS_ / 4 + 255) / 256), blk, 0, stream>>>(map);
    ppscatter_ids<<<dim3((npillars + 255) / 256), blk, 0, stream>>>(coords, map, npillars);
    ppscatter_gather<<<dim3(CELLS_ / 4 / 256), blk, 0, stream>>>(map, feat, out); // 837 blocks exact
}